// QuantumEntanglement_17849884082770
// MI455X (gfx1250) — compile-verified
//
#include <hip/hip_runtime.h>
#include <math.h>

// ---------------- problem constants ----------------
#define NTOK 1024     // B*T = 16*64
#define DD   128      // D
#define DIN  256
#define XD   40       // DTR + 2*DSTATE
#define DTR  8
#define CELL 64
#define NCLS 4

typedef __attribute__((ext_vector_type(2))) float v2f;
typedef __attribute__((ext_vector_type(8))) float v8f;

// CDNA5 f32 WMMA: D(16x16,f32) = A(16x4,f32) x B(4x16,f32) + C
__device__ __forceinline__ v8f wmma4(v2f a, v2f b, v8f c) {
  return __builtin_amdgcn_wmma_f32_16x16x4_f32(
      /*neg_a=*/false, a, /*neg_b=*/false, b,
      /*c_mod=*/(short)0, c, /*reuse_a=*/false, /*reuse_b=*/false);
}

__device__ __forceinline__ v8f zero8() {
  v8f z = {0.f, 0.f, 0.f, 0.f, 0.f, 0.f, 0.f, 0.f};
  return z;
}

// Fragment layout (wave32, from CDNA5 ISA):
//  A 16x4 f32 : lane%16 = M row; lanes>=16 hold K=2,3; vgpr j holds K = khalf+j
//  B 4x16 f32 : lane%16 = N col; same K striping
//  C/D 16x16  : vgpr r -> M = r + 8*(lane>=16); lane%16 = N
// One wave owns one 16x16 output tile, looping K in steps of 4.
__device__ __forceinline__ v8f gemm_lds_glb(const float* As, int lda,
                                            const float* __restrict__ Bg, int ldb,
                                            int nb, int K, int lane, v8f c) {
  const int mn = lane & 15;
  const int kh = (lane >> 4) << 1;
  for (int kb = 0; kb < K; kb += 4) {
    v2f a = *(const v2f*)(As + mn * lda + kb + kh);   // ds_load_b64
    v2f b;
    b.x = Bg[(size_t)(kb + kh + 0) * ldb + nb + mn];
    b.y = Bg[(size_t)(kb + kh + 1) * ldb + nb + mn];
    c = wmma4(a, b, c);
  }
  return c;
}

// ================= kernel 1: fused multimodal frontend =================
// reps = relu(x @ W + b) for text/audio/visual; phase = relu(beats@Wb+bb)
// h[d] = w0*(a0*a1)^2-style entanglement diagonal (only diagonal survives)
__global__ __launch_bounds__(256) void k_frontend(
    const float* __restrict__ text, const float* __restrict__ audio,
    const float* __restrict__ visual, const float* __restrict__ beats,
    const float* __restrict__ Wt, const float* __restrict__ bt,
    const float* __restrict__ Wa, const float* __restrict__ ba,
    const float* __restrict__ Wv, const float* __restrict__ bv,
    const float* __restrict__ Wb, const float* __restrict__ bb,
    float* __restrict__ h_out) {
  __shared__ float Ain[16 * 128];        // K-chunked input tile
  __shared__ float repS[4][16 * DD];     // rep0,rep1,rep2,phase
  __shared__ float nrmS[48];
  __shared__ float wS[48];

  const int tid = threadIdx.x, lane = tid & 31, wave = tid >> 5;
  const int tok0 = blockIdx.x * 16;
  const int nb = wave * 16;

  const float* ins[4] = {text, audio, visual, beats};
  const float* Wm[4] = {Wt, Wa, Wv, Wb};
  const float* bm[4] = {bt, ba, bv, bb};
  const int Ks[4] = {768, 128, 256, 768};

  if (tid == 0) { __builtin_prefetch(Wt, 0, 0); __builtin_prefetch(Wb, 0, 0); }

#pragma unroll
  for (int p = 0; p < 4; ++p) {
    const int K = Ks[p];
    const float* in = ins[p] + (size_t)tok0 * K;
    v8f c = zero8();
    for (int kc = 0; kc < K; kc += 128) {
      __syncthreads();  // protect Ain (previous consumer)
      for (int i = tid; i < 16 * 128; i += 256)
        Ain[i] = in[(size_t)(i >> 7) * K + kc + (i & 127)];
      __syncthreads();
      c = gemm_lds_glb(Ain, 128, Wm[p] + (size_t)kc * DD, DD, nb, 128, lane, c);
    }
    const int n = nb + (lane & 15);
    const int mo = (lane >> 4) * 8;
    const float bias = bm[p][n];
#pragma unroll
    for (int r = 0; r < 8; ++r)
      repS[p][(mo + r) * DD + n] = fmaxf(c[r] + bias, 0.f);
  }
  __syncthreads();

  // per-token L2 norms of the 3 reps
  if (tid < 48) {
    const int t = tid / 3, rp = tid % 3;
    float s = 0.f;
    for (int d = 0; d < DD; ++d) { float v = repS[rp][t * DD + d]; s += v * v; }
    nrmS[t * 3 + rp] = sqrtf(s);
  }
  __syncthreads();
  if (tid < 16) {  // softmax over 3 modal norms
    float a0 = nrmS[tid * 3 + 0], a1 = nrmS[tid * 3 + 1], a2 = nrmS[tid * 3 + 2];
    float mx = fmaxf(a0, fmaxf(a1, a2));
    float e0 = expf(a0 - mx), e1 = expf(a1 - mx), e2 = expf(a2 - mx);
    float inv = 1.f / (e0 + e1 + e2);
    wS[tid * 3 + 0] = e0 * inv; wS[tid * 3 + 1] = e1 * inv; wS[tid * 3 + 2] = e2 * inv;
  }
  __syncthreads();

  // entanglement diagonal -> h  (h_i is identically zero; dropped)
  for (int i = tid; i < 16 * DD; i += 256) {
    const int t = i >> 7;
    float n0 = fmaxf(nrmS[t * 3 + 0], 1e-12f);
    float n1 = fmaxf(nrmS[t * 3 + 1], 1e-12f);
    float n2 = fmaxf(nrmS[t * 3 + 2], 1e-12f);
    float a0 = repS[0][i] / n0, a1 = repS[1][i] / n1, a2 = repS[2][i] / n2;
    float ph = repS[3][i];
    float cp = cosf(ph), sp = sinf(ph);
    float r1 = a0 * cp, i1 = a0 * sp, r2 = a1 * cp, i2 = a1 * sp;
    float er = r1 * r2 - i1 * i2, ei = r1 * i2 + i1 * r2;
    float e0 = er * er + ei * ei;
    float r3 = a2 * cp, i3 = a2 * sp;
    er = r2 * r3 - i2 * i3; ei = r2 * i3 + i2 * r3;
    float e1 = er * er + ei * ei;
    h_out[(size_t)tok0 * DD + i] = wS[t * 3 + 0] * e0 + wS[t * 3 + 1] * e1;
  }
}

// ================= kernel 2: measurement-row normalization ==============
// urT[d][k] = meas_r[k][d] / ||meas[k,:]||  (transposed for coalesced B-loads)
__global__ __launch_bounds__(128) void k_meas_norm(
    const float* __restrict__ mr, const float* __restrict__ mi,
    float* __restrict__ urT, float* __restrict__ uiT) {
  __shared__ float red[128];
  const int k = blockIdx.x, d = threadIdx.x;
  const float a = mr[k * DD + d], b = mi[k * DD + d];
  red[d] = a * a + b * b;
  __syncthreads();
  for (int s = 64; s > 0; s >>= 1) {
    if (d < s) red[d] += red[d + s];
    __syncthreads();
  }
  const float inv = 1.f / sqrtf(red[0]);
  urT[d * DD + k] = a * inv;
  uiT[d * DD + k] = b * inv;
}

// ================= kernel 3: one fused Mamba layer ======================
__global__ __launch_bounds__(256) void k_mamba(
    const float* __restrict__ h_in, float* __restrict__ h_out,
    const float* __restrict__ W_in, const float* __restrict__ conv_w,
    const float* __restrict__ conv_b, const float* __restrict__ W_xproj,
    const float* __restrict__ W_dt, const float* __restrict__ b_dt,
    const float* __restrict__ Dp, const float* __restrict__ W_out) {
  __shared__ float hS[16 * DD];        // 8 KB
  __shared__ float xzS[16 * 512];      // 32 KB  (xin | z)
  __shared__ float xcS[16 * DIN];      // 16 KB  (xc, later reused as g)
  __shared__ float xdS[16 * 48];       // 3 KB   (dt | Bm | Cm, padded)
  __shared__ float sS[16];

  const int tid = threadIdx.x, lane = tid & 31, wave = tid >> 5;
  const int tok0 = blockIdx.x * 16;

  if (tid == 0) { __builtin_prefetch(W_in, 0, 0); __builtin_prefetch(W_out, 0, 0); }

  for (int i = tid; i < 16 * DD; i += 256) hS[i] = h_in[(size_t)tok0 * DD + i];
  __syncthreads();

  // xz = h @ W_in   [16,128]x[128,512]
  for (int tc = wave; tc < 32; tc += 8) {
    v8f c = zero8();
    c = gemm_lds_glb(hS, DD, W_in, 512, tc * 16, DD, lane, c);
    const int n = tc * 16 + (lane & 15);
    const int mo = (lane >> 4) * 8;
#pragma unroll
    for (int r = 0; r < 8; ++r) xzS[(mo + r) * 512 + n] = c[r];
  }
  __syncthreads();

  // xc = silu(xin * conv_w[:,3] + conv_b)
  for (int i = tid; i < 16 * DIN; i += 256) {
    const int t = i >> 8, j = i & 255;
    float v = xzS[t * 512 + j] * conv_w[j * 4 + 3] + conv_b[j];
    xcS[i] = v / (1.f + expf(-v));
  }
  __syncthreads();

  // xdbl = xc @ W_xproj   [16,256]x[256,40] (N padded to 48)
  if (wave < 3) {
    const int nb = wave * 16;
    const int col = nb + (lane & 15);
    const int kh = (lane >> 4) << 1;
    const int mrow = lane & 15;
    const bool ok = (col < XD);
    v8f c = zero8();
    for (int kb = 0; kb < DIN; kb += 4) {
      v2f a = *(const v2f*)(xcS + mrow * DIN + kb + kh);
      v2f b;
      b.x = ok ? W_xproj[(kb + kh + 0) * XD + col] : 0.f;
      b.y = ok ? W_xproj[(kb + kh + 1) * XD + col] : 0.f;
      c = wmma4(a, b, c);
    }
    const int mo = (lane >> 4) * 8;
    if (ok) {
#pragma unroll
      for (int r = 0; r < 8; ++r) xdS[(mo + r) * 48 + col] = c[r];
    }
  }
  __syncthreads();

  // s[t] = sum(Bm * Cm)
  if (tid < 16) {
    float s = 0.f;
    for (int j = 0; j < 16; ++j) s += xdS[tid * 48 + 8 + j] * xdS[tid * 48 + 24 + j];
    sS[tid] = s;
  }
  __syncthreads();

  // delta = softplus(dt @ W_dt + b_dt); g = (delta*xc*s + Dp*xc) * silu(z)
  for (int tc = wave; tc < 16; tc += 8) {
    v8f c = zero8();
    c = gemm_lds_glb(xdS, 48, W_dt, DIN, tc * 16, DTR, lane, c);
    const int n = tc * 16 + (lane & 15);
    const int mo = (lane >> 4) * 8;
#pragma unroll
    for (int r = 0; r < 8; ++r) {
      const int t = mo + r;
      float dv = c[r] + b_dt[n];
      float delta = (dv > 20.f) ? dv : log1pf(expf(dv));
      float xc = xcS[t * DIN + n];
      float y = delta * xc * sS[t] + Dp[n] * xc;
      float z = xzS[t * 512 + DIN + n];
      xcS[t * DIN + n] = y * (z / (1.f + expf(-z)));  // in-place: g overwrites xc
    }
  }
  __syncthreads();

  // h_out = g @ W_out  [16,256]x[256,128]
  {
    v8f c = zero8();
    const int nb = wave * 16;
    c = gemm_lds_glb(xcS, DIN, W_out, DD, nb, DIN, lane, c);
    const int n = nb + (lane & 15);
    const int mo = (lane >> 4) * 8;
#pragma unroll
    for (int r = 0; r < 8; ++r) h_out[(size_t)(tok0 + mo + r) * DD + n] = c[r];
  }
}

// ================= kernel 4: measurement + MLP head + log_softmax =======
__global__ __launch_bounds__(256) void k_head(
    const float* __restrict__ h, const float* __restrict__ urT,
    const float* __restrict__ uiT, const float* __restrict__ fc1_w,
    const float* __restrict__ fc1_b, const float* __restrict__ fc2_w,
    const float* __restrict__ fc2_b, float* __restrict__ out) {
  __shared__ float hS[16 * DD];
  __shared__ float pS[16 * DD];
  __shared__ float aS[16 * CELL];
  __shared__ float lS[16 * NCLS];

  const int tid = threadIdx.x, lane = tid & 31, wave = tid >> 5;
  const int tok0 = blockIdx.x * 16;

  for (int i = tid; i < 16 * DD; i += 256) hS[i] = h[(size_t)tok0 * DD + i];
  __syncthreads();

  // probs = (h@ur^T)^2 + (h@ui^T)^2 ; fused dual-accumulator WMMA loop
  {
    const int nb = wave * 16;
    const int mn = lane & 15;
    const int kh = (lane >> 4) << 1;
    v8f cr = zero8(), ci = zero8();
    for (int kb = 0; kb < DD; kb += 4) {
      v2f a = *(const v2f*)(hS + mn * DD + kb + kh);
      v2f br, bi;
      br.x = urT[(kb + kh + 0) * DD + nb + mn];
      br.y = urT[(kb + kh + 1) * DD + nb + mn];
      bi.x = uiT[(kb + kh + 0) * DD + nb + mn];
      bi.y = uiT[(kb + kh + 1) * DD + nb + mn];
      cr = wmma4(a, br, cr);
      ci = wmma4(a, bi, ci);
    }
    const int mo = (lane >> 4) * 8;
#pragma unroll
    for (int r = 0; r < 8; ++r) {
      float x = cr[r], y = ci[r];
      pS[(mo + r) * DD + nb + mn] = x * x + y * y;
    }
  }
  __syncthreads();

  // fc1 + relu   [16,128]x[128,64]
  if (wave < 4) {
    v8f c = zero8();
    c = gemm_lds_glb(pS, DD, fc1_w, CELL, wave * 16, DD, lane, c);
    const int n = wave * 16 + (lane & 15);
    const int mo = (lane >> 4) * 8;
#pragma unroll
    for (int r = 0; r < 8; ++r)
      aS[(mo + r) * CELL + n] = fmaxf(c[r] + fc1_b[n], 0.f);
  }
  __syncthreads();

  // fc2 + tanh (tiny: scalar dots)
  if (tid < 64) {
    const int t = tid >> 2, cc = tid & 3;
    float s = fc2_b[cc];
    for (int j = 0; j < CELL; ++j) s += aS[t * CELL + j] * fc2_w[j * NCLS + cc];
    lS[t * NCLS + cc] = tanhf(s);
  }
  __syncthreads();

  // log_softmax over 4 classes
  if (tid < 16) {
    float x0 = lS[tid * 4 + 0], x1 = lS[tid * 4 + 1];
    float x2 = lS[tid * 4 + 2], x3 = lS[tid * 4 + 3];
    float mx = fmaxf(fmaxf(x0, x1), fmaxf(x2, x3));
    float se = expf(x0 - mx) + expf(x1 - mx) + expf(x2 - mx) + expf(x3 - mx);
    float lse = mx + logf(se);
    float* o = out + (size_t)(tok0 + tid) * 4;
    o[0] = x0 - lse; o[1] = x1 - lse; o[2] = x2 - lse; o[3] = x3 - lse;
  }
}

// ======================== host-side launcher ============================
extern "C" void kernel_launch(void* const* d_in, const int* in_sizes, int n_in,
                              void* d_out, int out_size, void* d_ws, size_t ws_size,
                              hipStream_t stream) {
  const float* text   = (const float*)d_in[0];
  const float* audio  = (const float*)d_in[1];
  const float* visual = (const float*)d_in[2];
  const float* beats  = (const float*)d_in[3];
  const float* Wt = (const float*)d_in[4];  const float* bt = (const float*)d_in[5];
  const float* Wa = (const float*)d_in[6];  const float* ba = (const float*)d_in[7];
  const float* Wv = (const float*)d_in[8];  const float* bv = (const float*)d_in[9];
  const float* Wb = (const float*)d_in[10]; const float* bb = (const float*)d_in[11];
  const float* meas_r = (const float*)d_in[12];
  const float* meas_i = (const float*)d_in[13];
  const float* W_in   = (const float*)d_in[14];
  const float* conv_w = (const float*)d_in[15];
  const float* conv_b = (const float*)d_in[16];
  const float* W_xproj= (const float*)d_in[17];
  const float* W_dt   = (const float*)d_in[18];
  const float* b_dt   = (const float*)d_in[19];
  // d_in[20] = A_log (unused: seq_len==1 collapse)
  const float* Dp     = (const float*)d_in[21];
  const float* W_out  = (const float*)d_in[22];
  const float* fc1_w  = (const float*)d_in[23];
  const float* fc1_b  = (const float*)d_in[24];
  const float* fc2_w  = (const float*)d_in[25];
  const float* fc2_b  = (const float*)d_in[26];

  float* h0  = (float*)d_ws;
  float* h1  = h0 + (size_t)NTOK * DD;
  float* urT = h1 + (size_t)NTOK * DD;
  float* uiT = urT + (size_t)DD * DD;

  k_frontend<<<NTOK / 16, 256, 0, stream>>>(text, audio, visual, beats,
                                            Wt, bt, Wa, ba, Wv, bv, Wb, bb, h0);
  k_meas_norm<<<DD, 128, 0, stream>>>(meas_r, meas_i, urT, uiT);

  for (int l = 0; l < 2; ++l) {
    const float* src = (l == 0) ? h0 : h1;
    float* dst       = (l == 0) ? h1 : h0;
    k_mamba<<<NTOK / 16, 256, 0, stream>>>(
        src, dst,
        W_in    + (size_t)l * DD * 2 * DIN,
        conv_w  + (size_t)l * DIN * 4,
        conv_b  + (size_t)l * DIN,
        W_xproj + (size_t)l * DIN * XD,
        W_dt    + (size_t)l * DTR * DIN,
        b_dt    + (size_t)l * DIN,
        Dp      + (size_t)l * DIN,
        W_out   + (size_t)l * DIN * DD);
  }

  k_head<<<NTOK / 16, 256, 0, stream>>>(h0, urT, uiT, fc1_w, fc1_b,
                                        fc2_w, fc2_b, (float*)d_out);
}